// LocalCrossCorrelation2D_32384053412383
// MI455X (gfx1250) — compile-verified
//
#include <hip/hip_runtime.h>
#include <hip/hip_bf16.h>

typedef float v2f __attribute__((ext_vector_type(2)));
typedef float v4f __attribute__((ext_vector_type(4)));
typedef float v8f __attribute__((ext_vector_type(8)));

#define TILE    64
#define HALO    4
#define IN_DIM  72               // TILE + 2*HALO
#define STR     73               // odd (conflict-free row-strided access); 2*STR mod 64 = 18 in [16,48]
#define PLANE   (IN_DIM * STR)   // 5256 floats
#define HW      512
#define EPS_NCC 3.0590232050182579e-07f   // e^-15

// ---------------------------------------------------------------------------
// Kernel 1: one 64x64 tile of one batch image per workgroup (256 thr = 8 waves)
//   phase 1: zero-padded cooperative b128 load of 72x72 I/J halo region -> LDS
//   phase 2: horizontal 9-tap sliding box sums of {I, J, I^2, J^2, IJ} -> HQ
//   phase 3: vertical 9-tap box sums as banded-ones matmul on the matrix pipe
//            (v_wmma_f32_16x16x4_f32), then pointwise NCC math + reduction
// ---------------------------------------------------------------------------
__global__ __launch_bounds__(256) void ncc_tile_kernel(const float* __restrict__ I,
                                                       const float* __restrict__ J,
                                                       float* __restrict__ partials) {
  __shared__ float sI[PLANE];
  __shared__ float sJ[PLANE];
  __shared__ float HQ[5][PLANE];
  __shared__ float wsum[8];

  const int tid  = threadIdx.x;
  const int row0 = blockIdx.y * TILE;
  const int col0 = blockIdx.x * TILE;
  const float* __restrict__ Ib = I + (size_t)blockIdx.z * (HW * HW);
  const float* __restrict__ Jb = J + (size_t)blockIdx.z * (HW * HW);

  // ---------------- phase 1: vec4 load of halo region (zero pad outside image)
  // col0-4 is a multiple of 4 floats and the image edges are 16B-aligned, so
  // every float4 of the halo region is entirely in-bounds or entirely OOB.
  for (int t = tid; t < IN_DIM * (IN_DIM / 4); t += 256) {
    const int r  = t / (IN_DIM / 4);
    const int c4 = t - r * (IN_DIM / 4);
    const int gr = row0 - HALO + r;
    const int gc = col0 - HALO + c4 * 4;
    v4f vi = {0.f, 0.f, 0.f, 0.f};
    v4f vj = {0.f, 0.f, 0.f, 0.f};
    if (gr >= 0 && gr < HW && gc >= 0 && gc < HW) {
      const int gi = gr * HW + gc;
      vi = *reinterpret_cast<const v4f*>(Ib + gi);
      vj = *reinterpret_cast<const v4f*>(Jb + gi);
    }
    const int o = r * STR + c4 * 4;
    sI[o + 0] = vi.x; sI[o + 1] = vi.y; sI[o + 2] = vi.z; sI[o + 3] = vi.w;
    sJ[o + 0] = vj.x; sJ[o + 1] = vj.y; sJ[o + 2] = vj.z; sJ[o + 3] = vj.w;
  }
  __syncthreads();

  // ---------------- phase 2: horizontal 9-tap box sums (sliding window)
  // 288 tasks: (row r in [0,72), 16-col chunk q in [0,4))
  for (int task = tid; task < IN_DIM * 4; task += 256) {
    const int r  = task % IN_DIM;
    const int q  = task / IN_DIM;
    const float* __restrict__ rI = sI + r * STR;
    const float* __restrict__ rJ = sJ + r * STR;
    const int cb = q * 16;

    float hA = 0.0f, hB = 0.0f, hAA = 0.0f, hBB = 0.0f, hAB = 0.0f;
#pragma unroll
    for (int t9 = 0; t9 < 9; ++t9) {
      const float a = rI[cb + t9];
      const float b = rJ[cb + t9];
      hA += a; hB += b; hAA += a * a; hBB += b * b; hAB += a * b;
    }
    const int o = r * STR + cb;
    HQ[0][o] = hA; HQ[1][o] = hB; HQ[2][o] = hAA; HQ[3][o] = hBB; HQ[4][o] = hAB;

#pragma unroll
    for (int c = 1; c < 16; ++c) {
      const float an = rI[cb + c + 8];
      const float bn = rJ[cb + c + 8];
      const float ao = rI[cb + c - 1];
      const float bo = rJ[cb + c - 1];
      hA  += an - ao;
      hB  += bn - bo;
      hAA += an * an - ao * ao;
      hBB += bn * bn - bo * bo;
      hAB += an * bn - ao * bo;
      HQ[0][o + c] = hA; HQ[1][o + c] = hB; HQ[2][o + c] = hAA;
      HQ[3][o + c] = hBB; HQ[4][o + c] = hAB;
    }
  }
  __syncthreads();

  // ---------------- phase 3: vertical 9-tap via banded-ones WMMA (f32, K=4)
  // 16 sub-tiles of 16x16 (4 row-bands x 4 col-bands); wave w does 2 of them.
  const int wave = tid >> 5;
  const int lane = tid & 31;
  const int n    = lane & 15;   // B/D column; also A row m (m == lane&15)
  const int hi   = lane >> 4;   // half-wave select

  // Precompute the 6 banded-ones A chunks this lane contributes:
  // A layout (f32 16x4): lane holds row m=n; VGPR0 <-> K=4ch+2hi, VGPR1 <-> K+1
  float bandx[6], bandy[6];
#pragma unroll
  for (int ch = 0; ch < 6; ++ch) {
    const int k0 = 4 * ch + 2 * hi;
    bandx[ch] = (k0 >= n && k0 <= n + 8) ? 1.0f : 0.0f;
    bandy[ch] = (k0 + 1 >= n && k0 + 1 <= n + 8) ? 1.0f : 0.0f;
  }

  float part = 0.0f;

  for (int s = wave * 2; s < wave * 2 + 2; ++s) {
    const int R  = (s >> 2) << 4;   // sub-tile output row base in [0,64)
    const int c0 = (s & 3) << 4;    // sub-tile output col base in [0,64)

    v8f acc0 = {0.f,0.f,0.f,0.f,0.f,0.f,0.f,0.f};
    v8f acc1 = {0.f,0.f,0.f,0.f,0.f,0.f,0.f,0.f};
    v8f acc2 = {0.f,0.f,0.f,0.f,0.f,0.f,0.f,0.f};
    v8f acc3 = {0.f,0.f,0.f,0.f,0.f,0.f,0.f,0.f};
    v8f acc4 = {0.f,0.f,0.f,0.f,0.f,0.f,0.f,0.f};

#pragma unroll
    for (int ch = 0; ch < 6; ++ch) {
      const int k0 = 4 * ch + 2 * hi;
      v2f a;
      a.x = bandx[ch];
      a.y = bandy[ch];

      // B (4x16 chunk of HQ rows): VGPR0 <-> row k0, VGPR1 <-> row k0+1, col n
      const int base = (R + k0) * STR + c0 + n;
      v2f b;

      b.x = HQ[0][base]; b.y = HQ[0][base + STR];
      acc0 = __builtin_amdgcn_wmma_f32_16x16x4_f32(false, a, false, b, (short)0, acc0, false, false);
      b.x = HQ[1][base]; b.y = HQ[1][base + STR];
      acc1 = __builtin_amdgcn_wmma_f32_16x16x4_f32(false, a, false, b, (short)0, acc1, false, false);
      b.x = HQ[2][base]; b.y = HQ[2][base + STR];
      acc2 = __builtin_amdgcn_wmma_f32_16x16x4_f32(false, a, false, b, (short)0, acc2, false, false);
      b.x = HQ[3][base]; b.y = HQ[3][base + STR];
      acc3 = __builtin_amdgcn_wmma_f32_16x16x4_f32(false, a, false, b, (short)0, acc3, false, false);
      b.x = HQ[4][base]; b.y = HQ[4][base + STR];
      acc4 = __builtin_amdgcn_wmma_f32_16x16x4_f32(false, a, false, b, (short)0, acc4, false, false);
    }

    // pointwise NCC math on the 8 accumulator elements this lane owns
#pragma unroll
    for (int v = 0; v < 8; ++v) {
      const float SI  = acc0[v];
      const float SJ  = acc1[v];
      const float SII = acc2[v];
      const float SJJ = acc3[v];
      const float SIJ = acc4[v];

      const float uI = SI * (1.0f / 81.0f);
      const float uJ = SJ * (1.0f / 81.0f);

      float cross = SIJ - uJ * SI - uI * SJ + uI * uJ * 81.0f;
      float Iv    = SII - 2.0f * uI * SI + uI * uI * 81.0f;
      float Jv    = SJJ - 2.0f * uJ * SJ + uJ * uJ * 81.0f;

      const bool nz = (Iv * Jv) > EPS_NCC;
      const float cr = nz ? cross : 1.0f;
      const float iv = nz ? Iv : 1.0f;
      const float jv = nz ? Jv : 1.0f;

      part += (cr * cr) / (iv * jv + EPS_NCC);
    }
  }

  // ---------------- reduction: wave shuffle, then LDS across 8 waves
  part += __shfl_down(part, 16, 32);
  part += __shfl_down(part, 8, 32);
  part += __shfl_down(part, 4, 32);
  part += __shfl_down(part, 2, 32);
  part += __shfl_down(part, 1, 32);
  if (lane == 0) wsum[wave] = part;
  __syncthreads();
  if (tid == 0) {
    float tot = 0.0f;
#pragma unroll
    for (int w = 0; w < 8; ++w) tot += wsum[w];
    partials[(blockIdx.z * 8 + blockIdx.y) * 8 + blockIdx.x] = tot;
  }
}

// ---------------------------------------------------------------------------
// Kernel 2: per-batch finalize: out[b] = 1 - sum(partials[b, :64]) / 2^18
// ---------------------------------------------------------------------------
__global__ __launch_bounds__(64) void ncc_finalize(const float* __restrict__ partials,
                                                   float* __restrict__ out) {
  const int b = blockIdx.x;
  const int t = threadIdx.x;
  float v = partials[b * 64 + t];
  v += __shfl_down(v, 16, 32);
  v += __shfl_down(v, 8, 32);
  v += __shfl_down(v, 4, 32);
  v += __shfl_down(v, 2, 32);
  v += __shfl_down(v, 1, 32);
  __shared__ float s2[2];
  if ((t & 31) == 0) s2[t >> 5] = v;
  __syncthreads();
  if (t == 0) out[b] = 1.0f - (s2[0] + s2[1]) * (1.0f / 262144.0f);
}

extern "C" void kernel_launch(void* const* d_in, const int* in_sizes, int n_in,
                              void* d_out, int out_size, void* d_ws, size_t ws_size,
                              hipStream_t stream) {
  const float* I = (const float*)d_in[0];
  const float* J = (const float*)d_in[1];
  float* out      = (float*)d_out;
  float* partials = (float*)d_ws;   // 32*64 = 2048 floats, fully rewritten each launch

  dim3 grid(HW / TILE, HW / TILE, 32);   // 8 x 8 x 32
  ncc_tile_kernel<<<grid, 256, 0, stream>>>(I, J, partials);
  ncc_finalize<<<32, 64, 0, stream>>>(partials, out);
}